// FFEdgeCountingAutoencoder4_90443421319696
// MI455X (gfx1250) — compile-verified
//
#include <hip/hip_runtime.h>
#include <stdint.h>

// ---------------------------------------------------------------------------
// FFEdgeCountingAutoencoder forward on MI455X (gfx1250, wave32).
//
//   out[b,o] = op_o over i of (edge_sel(o,i) ? x[b,i] : identity(op_o)),
//   op_o in {min,max} chosen by gumbel-argmax over otc; edge_sel by
//   gumbel-argmax over etc[o, op_o, i, :].
//
// (min,max) is a tropical semiring -> v_wmma (D=A*B+C) is inapplicable.
// CDNA5-specific paths used instead:
//   * TENSOR_LOAD_TO_LDS (TDM): 2D strided x-tile (128x64, row stride IN)
//     DMA'd global->LDS with hardware padding (+4 dwords / 64 dwords) that
//     implements the bank-conflict-avoiding row stride of 68 for free.
//     Tracked with TENSORcnt, waited via __builtin_amdgcn_s_wait_tensorcnt.
//   * wave32 ballot to build edge-selection bitmasks (no LDS atomics/races).
//   * double-buffered LDS so TDM(c+1) + etc gumbel-argmax overlap the
//     masked min/max reduction of chunk c.
// ---------------------------------------------------------------------------

#define BATCH     128
#define CHUNK     64          // input columns staged in LDS per iteration
#define XSTRIDE   68          // CHUNK + 4 pad dwords (TDM pad_interval=64dw, pad_amount=4dw)
#define OTILE     4           // output neurons per block
#define NTHREADS  128         // 4 waves of 32; lane == batch row
#define NMASKW    (OTILE * (CHUNK / 32))

typedef uint32_t u32x4 __attribute__((ext_vector_type(4)));
typedef uint32_t u32x8 __attribute__((ext_vector_type(8)));

__device__ __forceinline__ uint32_t mix32(uint32_t x) {
  x ^= x >> 16; x *= 0x7feb352du;
  x ^= x >> 15; x *= 0x846ca68bu;
  x ^= x >> 16;
  return x;
}

// uniform in (0,1), then standard Gumbel
__device__ __forceinline__ float gumbel_from(uint32_t h) {
  float u = (float)(h >> 8) * 5.9604645e-08f + 2.9802322e-08f;
  return -__logf(-__logf(u));
}

// ---- Tensor Data Mover: load one 128 x 64 f32 tile (row stride IN) into LDS
// with +4-dword padding after every 64 dwords (-> LDS row stride 68 dwords).
// D# group layout per CDNA5 ISA.
__device__ __forceinline__ void tdm_load_tile(const float* gsrc,
                                              uint32_t lds_byte_addr, int IN) {
  const uint64_t ga = (uint64_t)(uintptr_t)gsrc;
  u32x4 g0;
  g0.x = 1u;                                        // count=1, user descriptor
  g0.y = lds_byte_addr;                             // lds_addr
  g0.z = (uint32_t)ga;                              // global_addr[31:0]
  g0.w = (uint32_t)((ga >> 32) & 0x1FFFFFFu)        // global_addr[56:32]
       | (2u << 30);                                // type=2 ("image")
  u32x8 g1;
  g1.s0 = (2u << 16)                                // data_size = 4B
        | (1u << 20)                                // pad_enable
        | (5u << 22)                                // pad_interval: 64 dwords
        | (3u << 25);                               // pad_amount: 4 dwords
  g1.s1 = ((uint32_t)IN & 0xFFFFu) << 16;           // tensor_dim0[15:0]
  g1.s2 = ((uint32_t)IN >> 16)                      // tensor_dim0[31:16]
        | ((uint32_t)BATCH << 16);                  // tensor_dim1[15:0]
  g1.s3 = ((uint32_t)CHUNK << 16);                  // tensor_dim1[31:16]=0 | tile_dim0
  g1.s4 = (uint32_t)BATCH;                          // tile_dim1 | tile_dim2=0
  g1.s5 = (uint32_t)IN;                             // tensor_dim0_stride[31:0]
  g1.s6 = 0u;                                       // stride[47:32] | dim1_stride lo
  g1.s7 = 0u;
  asm volatile("tensor_load_to_lds %0, %1" :: "s"(g0), "s"(g1) : "memory");
}

// ---- edge-type selection for one chunk: sel = argmax(etc[o,op,i,:]+Gumbel);
// each wave covers 32 consecutive i of one o -> ballot builds the mask word.
__device__ __forceinline__ void mask_gen(int c, uint32_t* mdst, int o0,
                                         uint32_t opbits,
                                         const float* __restrict__ etc,
                                         int IN, uint32_t seed, int t) {
  #pragma unroll
  for (int k = 0; k < (OTILE * CHUNK) / NTHREADS; ++k) {
    const int p  = k * NTHREADS + t;
    const int o  = p >> 6;                 // CHUNK == 64
    const int i  = p & 63;
    const int og = o0 + o;
    const int op = (int)((opbits >> o) & 1u);
    const int gi = c * CHUNK + i;
    const int eidx = ((og * 2 + op) * IN + gi) * 2;
    const float2 cc = *(const float2*)&etc[eidx];
    const uint32_t hb = mix32(seed ^ 0x68E31DA4u ^ (uint32_t)eidx);
    const float g0 = gumbel_from(hb);
    const float g1 = gumbel_from(mix32(hb ^ 0x1B56C4E9u));
    const uint32_t bal = __builtin_amdgcn_ballot_w32(cc.y + g1 > cc.x + g0);
    if ((t & 31) == 0)                     // lane 0 stores the wave's word
      mdst[o * (CHUNK / 32) + (i >> 5)] = bal;
  }
}

__global__ __launch_bounds__(NTHREADS)
void logic_layer_kernel(const float* __restrict__ xin,   // [BATCH, IN]
                        const float* __restrict__ etc,   // [OUT, 2, IN, 2]
                        const float* __restrict__ otc,   // [OUT, 2]
                        float* __restrict__ out,         // [BATCH, OUT]
                        int IN, int OUT, uint32_t seed)
{
  __shared__ float    xs[2][BATCH * XSTRIDE];   // 2 x 34,816 B (double buffer)
  __shared__ uint32_t maskw[2][NMASKW];
  __shared__ float    accsh[OTILE * BATCH];     // store-transpose staging

  const int t  = threadIdx.x;                   // == batch row in reduction
  const int o0 = blockIdx.x * OTILE;

  // ---- operator selection per output neuron: argmax(otc[o,:] + Gumbel) ----
  uint32_t opbits = 0u;                         // bit o: 1 -> max (T_Conorm)
  float accs[OTILE];
  #pragma unroll
  for (int o = 0; o < OTILE; ++o) {
    const int og = o0 + o;
    const float c0 = otc[og * 2 + 0];
    const float c1 = otc[og * 2 + 1];
    const float s0 = c0 + gumbel_from(mix32(seed ^ 0xB5297A4Du ^ (uint32_t)(og * 2 + 0)));
    const float s1 = c1 + gumbel_from(mix32(seed ^ 0xB5297A4Du ^ (uint32_t)(og * 2 + 1)));
    const uint32_t isMax = (s1 > s0) ? 1u : 0u;
    opbits |= isMax << o;
    accs[o] = isMax ? 0.0f : 1.0f;              // no_edge value == reduction identity
  }

  const int nChunks = IN / CHUNK;

  // ---- prologue: stage chunk 0 -------------------------------------------
  if (t < 32)
    tdm_load_tile(xin, (uint32_t)(uintptr_t)(&xs[0][0]), IN);
  mask_gen(0, maskw[0], o0, opbits, etc, IN, seed, t);
  if (t < 32)
    __builtin_amdgcn_s_wait_tensorcnt(0);
  __syncthreads();

  // ---- software-pipelined chunk loop -------------------------------------
  for (int c = 0; c < nChunks; ++c) {
    const int cur = c & 1, nxt = cur ^ 1;

    if (c + 1 < nChunks) {
      if (t < 32)
        tdm_load_tile(xin + (c + 1) * CHUNK,
                      (uint32_t)(uintptr_t)(&xs[nxt][0]), IN);
      mask_gen(c + 1, maskw[nxt], o0, opbits, etc, IN, seed, t);
      if (c + 2 < nChunks) {  // pull the following etc slice toward GL2
        const int nidx = ((o0 * 2 + (int)(opbits & 1u)) * IN + (c + 2) * CHUNK) * 2;
        __builtin_prefetch(&etc[nidx], 0, 0);
      }
    }

    // ---- masked tropical reduction on chunk c: lane == batch row ----------
    const float* rowp = &xs[cur][t * XSTRIDE];
    #pragma unroll
    for (int o = 0; o < OTILE; ++o) {
      float acc = accs[o];
      if (((opbits >> o) & 1u) == 0u) {          // T_Norm: min, unselected -> 1
        #pragma unroll
        for (int w = 0; w < CHUNK / 32; ++w) {
          const uint32_t m = maskw[cur][o * (CHUNK / 32) + w];
          #pragma unroll
          for (int k4 = 0; k4 < 8; ++k4) {
            const float4 v = *(const float4*)&rowp[w * 32 + k4 * 4];
            acc = fminf(acc, ((m >> (k4 * 4 + 0)) & 1u) ? v.x : 1.0f);
            acc = fminf(acc, ((m >> (k4 * 4 + 1)) & 1u) ? v.y : 1.0f);
            acc = fminf(acc, ((m >> (k4 * 4 + 2)) & 1u) ? v.z : 1.0f);
            acc = fminf(acc, ((m >> (k4 * 4 + 3)) & 1u) ? v.w : 1.0f);
          }
        }
      } else {                                   // T_Conorm: max, unselected -> 0
        #pragma unroll
        for (int w = 0; w < CHUNK / 32; ++w) {
          const uint32_t m = maskw[cur][o * (CHUNK / 32) + w];
          #pragma unroll
          for (int k4 = 0; k4 < 8; ++k4) {
            const float4 v = *(const float4*)&rowp[w * 32 + k4 * 4];
            acc = fmaxf(acc, ((m >> (k4 * 4 + 0)) & 1u) ? v.x : 0.0f);
            acc = fmaxf(acc, ((m >> (k4 * 4 + 1)) & 1u) ? v.y : 0.0f);
            acc = fmaxf(acc, ((m >> (k4 * 4 + 2)) & 1u) ? v.z : 0.0f);
            acc = fmaxf(acc, ((m >> (k4 * 4 + 3)) & 1u) ? v.w : 0.0f);
          }
        }
      }
      accs[o] = acc;
    }

    if (t < 32)
      __builtin_amdgcn_s_wait_tensorcnt(0);      // TDM(c+1) landed
    __syncthreads();                             // tile+masks visible, buffers free
  }

  // ---- coalesced epilogue store via LDS transpose ------------------------
  #pragma unroll
  for (int o = 0; o < OTILE; ++o) accsh[o * BATCH + t] = accs[o];
  __syncthreads();
  #pragma unroll
  for (int k = 0; k < OTILE; ++k) {
    const int f = k * NTHREADS + t;              // consecutive lanes -> ascending addrs
    const int b = f / OTILE;
    const int o = f % OTILE;
    out[(size_t)b * OUT + o0 + o] = accsh[o * BATCH + b];
  }
}

extern "C" void kernel_launch(void* const* d_in, const int* in_sizes, int n_in,
                              void* d_out, int out_size, void* d_ws, size_t ws_size,
                              hipStream_t stream) {
  (void)in_sizes; (void)n_in; (void)out_size; (void)ws_size;
  const float* x    = (const float*)d_in[0];
  const float* etc0 = (const float*)d_in[1];
  const float* otc0 = (const float*)d_in[2];
  const float* etc1 = (const float*)d_in[3];
  const float* otc1 = (const float*)d_in[4];
  float* out = (float*)d_out;
  float* h   = (float*)d_ws;                 // [128, 512] intermediate

  dim3 blk(NTHREADS);
  // Layer 0: 1024 -> 512   (128 blocks)
  logic_layer_kernel<<<dim3(512 / OTILE), blk, 0, stream>>>(
      x, etc0, otc0, h, 1024, 512, 0x9E3779B9u);
  // Layer 1: 512 -> 1024   (256 blocks)
  logic_layer_kernel<<<dim3(1024 / OTILE), blk, 0, stream>>>(
      h, etc1, otc1, out, 512, 1024, 0x85EBCA6Bu);
}